// Net_18966575579675
// MI455X (gfx1250) — compile-verified
//
#include <hip/hip_runtime.h>

typedef float v2f __attribute__((ext_vector_type(2)));
typedef float v8f __attribute__((ext_vector_type(8)));

#define BATCH   8192
#define NNET    722
#define CAMD    10
#define INDIM   11     // prior scalar + 10 camera dims
#define HID     10
#define LODN    6
#define NETS_PER_BLOCK  8
#define TILES_PER_BLOCK 8

// Load one 4x16 B-fragment (K x N) of a row-major [krows x ncols] weight
// matrix, zero-padded. Assumed B layout (mirror of documented A 16x4):
// lane L holds column N = L&15; VGPR v holds K = kbase + v + 2*(L>>4).
__device__ __forceinline__ v2f load_bfrag(const float* __restrict__ Wn,
                                          int krows, int ncols, int kbase,
                                          int lane) {
  const int n  = lane & 15;
  const int kh = (lane >> 4) << 1;
  const int k0 = kbase + kh;
  const int k1 = k0 + 1;
  v2f f;
  f.x = (k0 < krows && n < ncols) ? Wn[k0 * ncols + n] : 0.0f;
  f.y = (k1 < krows && n < ncols) ? Wn[k1 * ncols + n] : 0.0f;
  return f;
}

// Layer-1 input element x[b][k] for one network, branch-free:
// k==0 -> prior_lod[b][net]; 1..10 -> camera[b][k-1]; 11 -> 0 (pad).
// Address is clamped so every lane issues an unconditional load (no EXEC
// branch in the hot loop); pad lanes select 0 afterwards via cndmask.
__device__ __forceinline__ float in_elem(const float* __restrict__ prior,
                                         const float* __restrict__ camera,
                                         int b, int net, int k) {
  const int kc = (k <= CAMD) ? k : CAMD;              // clamp to valid range
  const float* p = (kc == 0) ? (prior + (size_t)b * NNET + net)
                             : (camera + (size_t)b * CAMD + (kc - 1));
  const float v = *p;                                  // unconditional load
  return (k <= CAMD) ? v : 0.0f;                       // zero the pad lane
}

__device__ __forceinline__ float relu1(float x) {
  return (x > 0.0f) ? x : 0.0f;                        // single v_max/cndmask
}

__global__ __launch_bounds__(256, 2) void lod_mlp_wmma(
    const float* __restrict__ prior, const float* __restrict__ camera,
    const float* __restrict__ W1, const float* __restrict__ b1,
    const float* __restrict__ W2, const float* __restrict__ b2,
    const float* __restrict__ W3, const float* __restrict__ b3,
    float* __restrict__ out)
{
  // one 16x16 f32 tile per wave: transpose buffer + output staging
  __shared__ float sh[NETS_PER_BLOCK * 256];

  const int lane  = threadIdx.x & 31;
  const int wave  = threadIdx.x >> 5;           // network within group
  const int group = blockIdx.y;                 // group of 8 networks
  const int net   = group * NETS_PER_BLOCK + wave;
  const int net_c = (net < NNET) ? net : (NNET - 1);   // clamp; stores predicated

  const int mrow   = lane & 15;        // A-fragment row (batch row in tile)
  const int khx2   = (lane >> 4) << 1; // A/B K offset of this half-wave (0|2)
  const int ncol   = lane & 15;        // B/C/D column
  const int rowOff = (lane >> 4) << 3; // D rows M = r (+8 for upper half)

  // ---- per-network weights -> registers in WMMA-B layout, zero padded ----
  const float* W1n = W1 + (size_t)net_c * INDIM * HID;
  const float* W2n = W2 + (size_t)net_c * HID * HID;
  const float* W3n = W3 + (size_t)net_c * HID * LODN;

  v2f w1f[3], w2f[3], w3f[3];
#pragma unroll
  for (int ks = 0; ks < 3; ++ks) {
    w1f[ks] = load_bfrag(W1n, INDIM, HID,  ks * 4, lane);
    w2f[ks] = load_bfrag(W2n, HID,   HID,  ks * 4, lane);
    w3f[ks] = load_bfrag(W3n, HID,   LODN, ks * 4, lane);
  }
  const float b1v = (ncol < HID)  ? b1[(size_t)net_c * HID  + ncol] : 0.0f;
  const float b2v = (ncol < HID)  ? b2[(size_t)net_c * HID  + ncol] : 0.0f;
  const float b3v = (ncol < LODN) ? b3[(size_t)net_c * LODN + ncol] : 0.0f;

  float* mysh = sh + wave * 256;
  const int tile0 = blockIdx.x * TILES_PER_BLOCK;

  for (int t = 0; t < TILES_PER_BLOCK; ++t) {
    const int b0   = (tile0 + t) * 16;
    const int brow = b0 + mrow;

    // ---------- layer 1: h1 = relu(x*W1 + b1), N padded to 16 ----------
    v8f acc;
#pragma unroll
    for (int j = 0; j < 8; ++j) acc[j] = b1v;   // bias folded into C
#pragma unroll
    for (int ks = 0; ks < 3; ++ks) {
      v2f a;
      a.x = in_elem(prior, camera, brow, net_c, ks * 4 + khx2);
      a.y = in_elem(prior, camera, brow, net_c, ks * 4 + khx2 + 1);
      acc = __builtin_amdgcn_wmma_f32_16x16x4_f32(false, a, false, w1f[ks],
                                                  (short)0, acc, false, false);
    }
#pragma unroll
    for (int j = 0; j < 8; ++j) acc[j] = relu1(acc[j]);

    // ---- D->A transpose through LDS (in-order within wave) ----
    __builtin_amdgcn_wave_barrier();
#pragma unroll
    for (int r = 0; r < 8; ++r) mysh[(r + rowOff) * 16 + ncol] = acc[r];
    __builtin_amdgcn_wave_barrier();

    // ---------- layer 2: h2 = relu(h1*W2 + b2) ----------
    v8f acc2;
#pragma unroll
    for (int j = 0; j < 8; ++j) acc2[j] = b2v;
#pragma unroll
    for (int ks = 0; ks < 3; ++ks) {
      v2f a = *(const v2f*)&mysh[mrow * 16 + ks * 4 + khx2]; // ds_load_b64
      acc2 = __builtin_amdgcn_wmma_f32_16x16x4_f32(false, a, false, w2f[ks],
                                                   (short)0, acc2, false, false);
    }
#pragma unroll
    for (int j = 0; j < 8; ++j) acc2[j] = relu1(acc2[j]);

    __builtin_amdgcn_wave_barrier();
#pragma unroll
    for (int r = 0; r < 8; ++r) mysh[(r + rowOff) * 16 + ncol] = acc2[r];
    __builtin_amdgcn_wave_barrier();

    // ---------- layer 3: out = h2*W3 + b3 ----------
    v8f acc3;
#pragma unroll
    for (int j = 0; j < 8; ++j) acc3[j] = b3v;
#pragma unroll
    for (int ks = 0; ks < 3; ++ks) {
      v2f a = *(const v2f*)&mysh[mrow * 16 + ks * 4 + khx2];
      acc3 = __builtin_amdgcn_wmma_f32_16x16x4_f32(false, a, false, w3f[ks],
                                                   (short)0, acc3, false, false);
    }

    // ---- stage D3 into LDS, then block-wide coalesced store ----
#pragma unroll
    for (int r = 0; r < 8; ++r) mysh[(r + rowOff) * 16 + ncol] = acc3[r];
    __syncthreads();

    // 16 rows x 8 nets x 6 lods = 768 floats; contiguous 192B runs per row
#pragma unroll
    for (int i = 0; i < 3; ++i) {
      const int e  = (int)threadIdx.x + i * 256;  // 0..767
      const int m  = e / 48;
      const int rm = e - m * 48;
      const int w  = rm / 6;
      const int o  = rm - w * 6;
      const int nn = group * NETS_PER_BLOCK + w;
      if (nn < NNET)
        out[((size_t)(b0 + m) * NNET + nn) * LODN + o] = sh[w * 256 + m * 16 + o];
    }
    __syncthreads();   // guard LDS reuse in next tile iteration
  }
}

extern "C" void kernel_launch(void* const* d_in, const int* in_sizes, int n_in,
                              void* d_out, int out_size, void* d_ws, size_t ws_size,
                              hipStream_t stream) {
  (void)in_sizes; (void)n_in; (void)d_ws; (void)ws_size; (void)out_size;
  const float* prior  = (const float*)d_in[0];
  const float* camera = (const float*)d_in[1];
  const float* W1 = (const float*)d_in[2];
  const float* b1 = (const float*)d_in[3];
  const float* W2 = (const float*)d_in[4];
  const float* b2 = (const float*)d_in[5];
  const float* W3 = (const float*)d_in[6];
  const float* b3 = (const float*)d_in[7];
  float* out = (float*)d_out;

  dim3 grid(BATCH / 16 / TILES_PER_BLOCK,                 // 64 batch-tile groups
            (NNET + NETS_PER_BLOCK - 1) / NETS_PER_BLOCK); // 91 network groups
  lod_mlp_wmma<<<grid, 256, 0, stream>>>(prior, camera, W1, b1, W2, b2, W3, b3, out);
}